// BGNLLLoss_12902081757779
// MI455X (gfx1250) — compile-verified
//
#include <hip/hip_runtime.h>

typedef __attribute__((ext_vector_type(2))) float v2f;
typedef __attribute__((ext_vector_type(8))) float v8f;

#define TT 4096
#define PP 4096
#define CT 256                 // T rows per chunk
#define NCHUNK (TT / CT)       // 16 chunks -> 16*4096*4 = 256 KB workspace
#define COLS_PER_BLOCK 128     // 8 waves * 16 columns
#define THREADS 256

// Hardware transcendental wrappers (guarantee v_exp_f32 / v_log_f32 / v_rcp_f32,
// avoiding IEEE-division expansion and libm calls).
__device__ __forceinline__ float hw_exp(float x) {          // e^x
    return __builtin_amdgcn_exp2f(x * 1.44269504088896340736f);
}
__device__ __forceinline__ float hw_log2(float x) {         // log2(x)
    return __builtin_amdgcn_logf(x);
}
__device__ __forceinline__ float hw_rcp(float x) {          // 1/x
    return __builtin_amdgcn_rcpf(x);
}

// Per-element negative log-likelihood of the bivariate Gaussian, in log
// space.  log(sx*sy) = lsx + lsy exactly (params store log-sigma).
// nll = -log(max(pdf, 1e-20)) = min(-logpdf, -log(1e-20)).
__device__ __forceinline__ float nll_elem(float tx, float ty,
                                          float mux, float muy,
                                          float lsx, float lsy, float pc) {
    // corr = tanh(pc) = 1 - 2/(e^{2x}+1)   (saturates correctly at +-1)
    float e2   = hw_exp(2.0f * pc);
    float corr = 1.0f - 2.0f * hw_rcp(e2 + 1.0f);
    float isx  = hw_exp(-lsx);
    float isy  = hw_exp(-lsy);
    float nx = tx - mux;
    float ny = ty - muy;
    float ax = nx * isx;
    float ay = ny * isy;
    float z  = ax * ax + ay * ay - 2.0f * corr * ax * ay;
    float nr = 1.0f - corr * corr;                      // 1 - rho^2
    const float LOG_2PI    = 1.8378770664093453f;       // ln(2*pi)
    const float NEG_HALFLN2 = -0.34657359027997264f;    // -ln(2)/2
    float logpdf = -(0.5f * z) * hw_rcp(nr) - LOG_2PI - (lsx + lsy)
                   + NEG_HALFLN2 * hw_log2(nr);         // == -0.5*ln(nr)
    return fminf(-logpdf, 46.051701859880914f);         // clip(pdf, 1e-20)
}

// Phase 1: each wave reduces a CT x 16 tile of nll values into 16 column
// sums with V_WMMA_F32_16X16X4_F32, A = ones (exact f32 adds, K-layout
// permutation invariant).  B 4x16: lanes 0-15 hold K=0,1; lanes 16-31 K=2,3;
// column n = lane & 15.
__global__ void __launch_bounds__(THREADS)
bgnll_partial_kernel(const float* __restrict__ targets,
                     const float* __restrict__ params,
                     float* __restrict__ partial) {
    const int lane = threadIdx.x & 31;
    const int wave = threadIdx.x >> 5;
    const int n    = lane & 15;
    const int kb   = (lane >> 4) << 1;                  // 0 or 2
    const int p    = blockIdx.x * COLS_PER_BLOCK + wave * 16 + n;
    const int t0   = blockIdx.y * CT;

    v2f a; a.x = 1.0f; a.y = 1.0f;                      // ones A-matrix
    v8f c = {};                                         // accumulator

#pragma unroll 4
    for (int t = t0; t < t0 + CT; t += 4) {
        const int ia = (t + kb) * PP + p;               // element (t+kb,   p)
        const int ib = ia + PP;                         // element (t+kb+1, p)

        const float2 g0 = *(const float2*)(targets + 2 * ia);
        const float2 g1 = *(const float2*)(targets + 2 * ib);
        const float* pr0 = params + 5 * ia;
        const float* pr1 = params + 5 * ib;

        v2f b;
        b.x = nll_elem(g0.x, g0.y, pr0[0], pr0[1], pr0[2], pr0[3], pr0[4]);
        b.y = nll_elem(g1.x, g1.y, pr1[0], pr1[1], pr1[2], pr1[3], pr1[4]);

        // D = ones(16x4) * B(4x16) + C : every row of D = column sums.
        // C-RAW accumulate chain is the legal pipelined WMMA pattern.
        c = __builtin_amdgcn_wmma_f32_16x16x4_f32(
                /*neg_a=*/false, a, /*neg_b=*/false, b,
                /*c_mod=*/(short)0, c, /*reuse_a=*/false, /*reuse_b=*/false);
    }

    // All D rows hold identical column sums; c[0] on lanes 0-15 is
    // colsum[N=lane] (M=0 row); lanes 16-31 would duplicate (M=8 row).
    if (lane < 16) partial[blockIdx.y * PP + p] = c[0];
}

// Phase 2: deterministic fixed-order reduction of the 16 chunk partials.
__global__ void bgnll_reduce_kernel(const float* __restrict__ partial,
                                    float* __restrict__ out) {
    const int p = blockIdx.x * blockDim.x + threadIdx.x;
    if (p < PP) {
        float s = 0.0f;
        for (int ch = 0; ch < NCHUNK; ++ch)
            s += partial[ch * PP + p];
        out[p] = s;                                     // overwrite, no init needed
    }
}

extern "C" void kernel_launch(void* const* d_in, const int* in_sizes, int n_in,
                              void* d_out, int out_size, void* d_ws, size_t ws_size,
                              hipStream_t stream) {
    const float* targets = (const float*)d_in[0];   // [T, P, 2] f32
    const float* params  = (const float*)d_in[1];   // [T, P, 5] f32
    // d_in[2] (peopleIDs) is an identity arange; reference ignores it.
    float* out     = (float*)d_out;                 // [P] f32
    float* partial = (float*)d_ws;                  // NCHUNK * P floats (256 KB)

    dim3 grid(PP / COLS_PER_BLOCK, NCHUNK);         // 32 x 16 = 512 blocks
    bgnll_partial_kernel<<<grid, THREADS, 0, stream>>>(targets, params, partial);
    bgnll_reduce_kernel<<<(PP + 255) / 256, 256, 0, stream>>>(partial, out);
}